// RBFN_11484742549553
// MI455X (gfx1250) — compile-verified
//
#include <hip/hip_runtime.h>
#include <hip/hip_bf16.h>

// Problem constants (from reference)
#define NROWS 131072
#define DDIM  32
#define CC    512
#define DOUT  3

typedef __attribute__((ext_vector_type(2))) float v2f;
typedef __attribute__((ext_vector_type(8))) float v8f;

// LDS layout (floats). Centers padded to stride 34 words:
//  - even stride -> all row offsets (row*34 + 2*half + 4*s) are even words
//    => 8-byte aligned ds_load_b64
//  - 34 = 2*17, 17 coprime to 32 => 16 lanes hit 16 distinct banks (no conflicts)
#define CEN_STRIDE 34
#define SM_CEN   0
#define SM_CSQ   (CC * CEN_STRIDE)          // 17408
#define SM_K     (SM_CSQ + CC)              // 17920
#define SM_W     (SM_K + CC)                // 18432
#define SM_FLOATS (SM_W + DOUT * CC)        // 19968 floats = 79872 bytes

__device__ __forceinline__ float fast_exp2(float x) {
#if __has_builtin(__builtin_amdgcn_exp2f)
    return __builtin_amdgcn_exp2f(x);
#else
    return exp2f(x);
#endif
}

extern "C" __global__ __launch_bounds__(256)
void rbfn_wmma_kernel(const float* __restrict__ X,    // [N, 32]
                      const float* __restrict__ Cen,  // [512, 32]
                      const float* __restrict__ Sig,  // [512]
                      const float* __restrict__ W,    // [3, 512]
                      const float* __restrict__ Bias, // [3]
                      float* __restrict__ Out)        // [N, 3]
{
    extern __shared__ float sm[];
    float* smc   = sm + SM_CEN;
    float* smcsq = sm + SM_CSQ;
    float* smk   = sm + SM_K;
    float* smw   = sm + SM_W;

    const int tid = threadIdx.x;

    // ---- Block prologue: stage centers (padded), W into LDS ----
    for (int i4 = tid; i4 < (CC * DDIM / 4); i4 += 256) {   // 4096 float4's
        const int c   = i4 >> 3;            // 8 float4 per 32-float row
        const int off = (i4 & 7) << 2;
        const float4 v = ((const float4*)Cen)[i4];
        float* dst = smc + c * CEN_STRIDE + off;
        dst[0] = v.x; dst[1] = v.y; dst[2] = v.z; dst[3] = v.w;
    }
    for (int i = tid; i < DOUT * CC; i += 256) smw[i] = W[i];
    __syncthreads();

    // ---- Per-center constants: ||c||^2 and k = -log2(e)/sigma^2 ----
    for (int c = tid; c < CC; c += 256) {
        const float* row = smc + c * CEN_STRIDE;
        float s = 0.0f;
#pragma unroll
        for (int kk = 0; kk < DDIM; ++kk) s = fmaf(row[kk], row[kk], s);
        smcsq[c] = s;
        const float sg = Sig[c];
        smk[c] = -1.4426950408889634f / (sg * sg);
    }
    __syncthreads();

    // ---- Per-wave 16-row tile ----
    const int lane = tid & 31;
    const int wave = tid >> 5;
    const int mcol = lane & 15;   // A-row / B-column index within the 16x16 tile
    const int half = lane >> 4;   // K-pair select (per WMMA f32 16x16x4 layout)
    const int row0 = (blockIdx.x * 8 + wave) * 16;

    // A registers: lane holds batches[row0+mcol, 4s + 2*half + {0,1}] for s=0..7
    v2f areg[8];
    const float* xrow = X + (row0 + mcol) * DDIM + 2 * half;
#pragma unroll
    for (int s = 0; s < 8; ++s)
        areg[s] = *(const v2f*)(xrow + 4 * s);

    // ||x||^2 for row (row0+mcol): own 16 K-values + partner half via shfl_xor(16)
    float xp = 0.0f;
#pragma unroll
    for (int s = 0; s < 8; ++s)
        xp = fmaf(areg[s].x, areg[s].x, fmaf(areg[s].y, areg[s].y, xp));
    xp += __shfl_xor(xp, 16, 32);

    // Redistribute: accumulator row r of this lane corresponds to M = r + 8*half,
    // whose ||x||^2 lives in lane (8*half + r).
    float xsq[8];
#pragma unroll
    for (int r = 0; r < 8; ++r)
        xsq[r] = __shfl(xp, half * 8 + r, 32);

    float acc0[8], acc1[8], acc2[8];
#pragma unroll
    for (int r = 0; r < 8; ++r) { acc0[r] = 0.0f; acc1[r] = 0.0f; acc2[r] = 0.0f; }

    // ---- Main loop over 32 center tiles of 16 ----
    for (int ct = 0; ct < CC / 16; ++ct) {
        const int col = ct * 16 + mcol;

        // B registers from LDS (8-byte aligned, bank-conflict-free)
        const float* brow = smc + col * CEN_STRIDE + 2 * half;
        v2f breg[8];
#pragma unroll
        for (int s = 0; s < 8; ++s)
            breg[s] = *(const v2f*)(brow + 4 * s);

        // cross = X_tile @ Cen_tile^T via 8 chained f32 WMMAs (K=32)
        v8f cacc = {};
#pragma unroll
        for (int s = 0; s < 8; ++s)
            cacc = __builtin_amdgcn_wmma_f32_16x16x4_f32(
                false, areg[s], false, breg[s], (short)0, cacc, false, false);

        const float k   = smk[col];
        const float kc  = k * smcsq[col];
        const float n2k = -2.0f * k;
        const float w0 = smw[col], w1 = smw[CC + col], w2 = smw[2 * CC + col];

#pragma unroll
        for (int r = 0; r < 8; ++r) {
            const float t   = fmaf(k, xsq[r], kc);          // k*(x^2 + c^2)
            const float arg = fmaf(n2k, cacc[r], t);        // k*(x^2 + c^2 - 2*cross)
            const float e   = fast_exp2(arg);               // exp(-dist2/sigma^2)
            acc0[r] = fmaf(e, w0, acc0[r]);
            acc1[r] = fmaf(e, w1, acc1[r]);
            acc2[r] = fmaf(e, w2, acc2[r]);
        }
    }

    // ---- Reduce over the 16 columns within each half-wave ----
#pragma unroll
    for (int mask = 1; mask <= 8; mask <<= 1) {
#pragma unroll
        for (int r = 0; r < 8; ++r) {
            acc0[r] += __shfl_xor(acc0[r], mask, 32);
            acc1[r] += __shfl_xor(acc1[r], mask, 32);
            acc2[r] += __shfl_xor(acc2[r], mask, 32);
        }
    }

    // Lanes 0 and 16 hold the complete sums for rows [row0..row0+7] / [row0+8..+15]
    if (mcol == 0) {
        const float b0 = Bias[0], b1 = Bias[1], b2 = Bias[2];
#pragma unroll
        for (int r = 0; r < 8; ++r) {
            const int row = row0 + half * 8 + r;
            float* o = Out + row * DOUT;
            o[0] = acc0[r] + b0;
            o[1] = acc1[r] + b1;
            o[2] = acc2[r] + b2;
        }
    }
}

extern "C" void kernel_launch(void* const* d_in, const int* in_sizes, int n_in,
                              void* d_out, int out_size, void* d_ws, size_t ws_size,
                              hipStream_t stream) {
    (void)in_sizes; (void)n_in; (void)out_size; (void)d_ws; (void)ws_size;
    const float* X   = (const float*)d_in[0];
    const float* Cen = (const float*)d_in[1];
    const float* Sig = (const float*)d_in[2];
    const float* W   = (const float*)d_in[3];
    const float* B   = (const float*)d_in[4];
    float* Out = (float*)d_out;

    const size_t smem = (size_t)SM_FLOATS * sizeof(float);  // 79,872 bytes
    // gfx1250 WGPs have 320KB LDS; raise the per-kernel dynamic-LDS cap.
    hipFuncSetAttribute((const void*)rbfn_wmma_kernel,
                        hipFuncAttributeMaxDynamicSharedMemorySize, (int)smem);

    dim3 grid(NROWS / 128);   // 8 waves/block * 16 rows/wave = 128 rows/block
    dim3 block(256);
    rbfn_wmma_kernel<<<grid, block, smem, stream>>>(X, Cen, Sig, W, B, Out);
}